// SequenceExtract_77953656423028
// MI455X (gfx1250) — compile-verified
//
#include <hip/hip_runtime.h>
#include <hip/hip_bf16.h>

// Problem shape (from reference setup_inputs)
constexpr int Bb = 8;
constexpr int Ss = 4096;
constexpr int Hh = 1024;
constexpr int TS_M = 32;            // compacted rows per workgroup in the GEMM
constexpr int A_STRIDE = Hh + 4;    // padded LDS row stride (floats) -> conflict-free
constexpr float KEEP_RATIO = 0.8f;

typedef __attribute__((ext_vector_type(2))) float v2f;
typedef __attribute__((ext_vector_type(8))) float v8f;

// ---------------------------------------------------------------------------
// amin = min(attention_mask) (global), bmax = max(bias)
// ---------------------------------------------------------------------------
__global__ __launch_bounds__(256) void reduce_minmax_kernel(
    const float* __restrict__ amask, const float* __restrict__ bias,
    float* __restrict__ ws_f) {
  __shared__ float red[256];
  const int t = threadIdx.x;
  float mn = 3.0e38f;
  for (int i = t; i < Bb * Ss; i += 256) mn = fminf(mn, amask[i]);
  red[t] = mn;
  __syncthreads();
  for (int off = 128; off > 0; off >>= 1) {
    if (t < off) red[t] = fminf(red[t], red[t + off]);
    __syncthreads();
  }
  const float amin = red[0];
  __syncthreads();
  float mx = -3.0e38f;
  for (int i = t; i < Hh; i += 256) mx = fmaxf(mx, bias[i]);
  red[t] = mx;
  __syncthreads();
  for (int off = 128; off > 0; off >>= 1) {
    if (t < off) red[t] = fmaxf(red[t], red[t + off]);
    __syncthreads();
  }
  if (t == 0) { ws_f[0] = amin; ws_f[1] = red[0]; }
}

// ---------------------------------------------------------------------------
// Per-batch stream compaction: comp_idx[b][j] = original index of j-th kept
// row (previous_mask > 0.5), cnt[b] = number kept. Block scan, 1024 threads.
// ---------------------------------------------------------------------------
__global__ __launch_bounds__(1024) void compact_kernel(
    const float* __restrict__ prev, int* __restrict__ comp,
    int* __restrict__ cnt) {
  __shared__ int partial[1024];
  const int b = blockIdx.x;
  const int t = threadIdx.x;
  const int base = t * 4;
  int flags[4];
  int local = 0;
  for (int i = 0; i < 4; ++i) {
    flags[i] = (prev[b * Ss + base + i] > 0.5f) ? 1 : 0;
    local += flags[i];
  }
  partial[t] = local;
  __syncthreads();
  for (int off = 1; off < 1024; off <<= 1) {
    int add = (t >= off) ? partial[t - off] : 0;
    __syncthreads();
    partial[t] += add;
    __syncthreads();
  }
  int pos = partial[t] - local;  // exclusive prefix
  for (int i = 0; i < 4; ++i) {
    if (flags[i]) comp[b * Ss + pos++] = base + i;
  }
  if (t == 1023) cnt[b] = partial[1023];
}

// ---------------------------------------------------------------------------
// new_ret == hidden_states (proved from _unextract semantics): pure copy.
// ---------------------------------------------------------------------------
__global__ __launch_bounds__(256) void copy_kernel(
    const float4* __restrict__ src, float4* __restrict__ dst, int n4) {
  const int i = blockIdx.x * 256 + threadIdx.x;
  if (i < n4) dst[i] = src[i];
}

// ---------------------------------------------------------------------------
// scores[b][j] = max_k( h[b, comp[j], :] . W[:, k] + bias[k] )  for j < cnt[b]
//
// f32 WMMA GEMM (V_WMMA_F32_16X16X4_F32), register-blocked 2x2:
// each wave computes BOTH 16-row M-tiles against an N-tile pair, so every
// A fragment and every B fragment feeds two WMMAs -> per-WMMA load cost is
// 1 global b32 + 0.5 LDS b64 (vs 2 + 0.5 unblocked).  K = 1024 in steps of 4.
// Row-max reduced across lanes with __shfl_xor (wave32, 16-lane halves per
// the C/D VGPR layout), then combined across the 8 waves in LDS.
// ---------------------------------------------------------------------------
__global__ __launch_bounds__(256) void gemm_scores_kernel(
    const float* __restrict__ hs, const float* __restrict__ Wm,
    const float* __restrict__ bias, const int* __restrict__ comp,
    const int* __restrict__ cnt, float* __restrict__ scores) {
  extern __shared__ float lds[];
  float* A_lds = lds;                    // TS_M * A_STRIDE floats
  float* part  = lds + TS_M * A_STRIDE;  // 8 waves * 2 mtiles * 16 rows

  const int b    = blockIdx.y;
  const int j0   = blockIdx.x * TS_M;
  const int cntb = cnt[b];
  if (j0 >= cntb) return;  // uniform across block: safe before barriers

  const int tid = threadIdx.x;

  // Stage A tile: 32 rows x 1024, gathered through comp_idx; pad rows = 0.
  {
    const int r = tid >> 3;  // 0..31 row
    const int q = tid & 7;   // column eighth
    const int j = j0 + r;
    const float4* src4 = nullptr;
    if (j < cntb) {
      const int orig = comp[b * Ss + j];
      src4 = (const float4*)(hs + ((size_t)b * Ss + orig) * Hh);
    }
    float4* dst4 = (float4*)(A_lds + r * A_STRIDE);
    const float4 z = make_float4(0.f, 0.f, 0.f, 0.f);
#pragma unroll 4
    for (int i = 0; i < 32; ++i) {
      const int c4 = q * 32 + i;
      dst4[c4] = src4 ? src4[c4] : z;
    }
  }
  __syncthreads();

  const int w    = tid >> 5;
  const int lane = tid & 31;
  const int hi   = lane >> 4;  // half-wave: K offset 0 or 2 (ISA A layout)
  const int ln   = lane & 15;
  const float* Arow0 = A_lds + ln * A_STRIDE;         // M-tile 0: rows 0..15
  const float* Arow1 = A_lds + (16 + ln) * A_STRIDE;  // M-tile 1: rows 16..31

  float rmax0[8], rmax1[8];
#pragma unroll
  for (int r = 0; r < 8; ++r) { rmax0[r] = -3.0e38f; rmax1[r] = -3.0e38f; }

  for (int t = 0; t < 4; ++t) {
    const int p = w + 8 * t;  // N-tile pair 0..31 (unique per wave)
    const int colA = p * 32 + ln;
    const int colB = colA + 16;
    v8f c00, c01, c10, c11;
#pragma unroll
    for (int r = 0; r < 8; ++r) {
      c00[r] = 0.f; c01[r] = 0.f; c10[r] = 0.f; c11[r] = 0.f;
    }
#pragma unroll 4
    for (int kb = 0; kb < Hh; kb += 4) {
      const int k0 = kb + (hi << 1);
      const v2f a0 = *(const v2f*)(Arow0 + k0);  // K = k0, k0+1 (this half)
      const v2f a1 = *(const v2f*)(Arow1 + k0);
      const float* wk = Wm + (size_t)k0 * Hh;
      v2f b0, b1;
      b0.x = wk[colA];      b0.y = wk[Hh + colA];
      b1.x = wk[colB];      b1.y = wk[Hh + colB];
      c00 = __builtin_amdgcn_wmma_f32_16x16x4_f32(false, a0, false, b0,
                                                  (short)0, c00, false, false);
      c01 = __builtin_amdgcn_wmma_f32_16x16x4_f32(false, a0, false, b1,
                                                  (short)0, c01, false, false);
      c10 = __builtin_amdgcn_wmma_f32_16x16x4_f32(false, a1, false, b0,
                                                  (short)0, c10, false, false);
      c11 = __builtin_amdgcn_wmma_f32_16x16x4_f32(false, a1, false, b1,
                                                  (short)0, c11, false, false);
    }
    const float biasA = bias[colA];
    const float biasB = bias[colB];
#pragma unroll
    for (int r = 0; r < 8; ++r) {
      // C layout: VGPR r holds row (r + 8*hi) of the tile at column = ln
      float m0 = fmaxf(c00[r] + biasA, c01[r] + biasB);
      m0 = fmaxf(m0, __shfl_xor(m0, 1, 16));
      m0 = fmaxf(m0, __shfl_xor(m0, 2, 16));
      m0 = fmaxf(m0, __shfl_xor(m0, 4, 16));
      m0 = fmaxf(m0, __shfl_xor(m0, 8, 16));
      rmax0[r] = fmaxf(rmax0[r], m0);
      float m1 = fmaxf(c10[r] + biasA, c11[r] + biasB);
      m1 = fmaxf(m1, __shfl_xor(m1, 1, 16));
      m1 = fmaxf(m1, __shfl_xor(m1, 2, 16));
      m1 = fmaxf(m1, __shfl_xor(m1, 4, 16));
      m1 = fmaxf(m1, __shfl_xor(m1, 8, 16));
      rmax1[r] = fmaxf(rmax1[r], m1);
    }
  }

  if (ln == 0) {  // lanes 0 and 16 of each wave
#pragma unroll
    for (int r = 0; r < 8; ++r) {
      part[(w * 2 + 0) * 16 + hi * 8 + r] = rmax0[r];
      part[(w * 2 + 1) * 16 + hi * 8 + r] = rmax1[r];
    }
  }
  __syncthreads();

  if (tid < 32) {  // combine 8 wave partials per (mtile,row), write scores
    const int m = tid >> 4;
    const int row = tid & 15;
    float v = -3.0e38f;
#pragma unroll
    for (int ww = 0; ww < 8; ++ww) v = fmaxf(v, part[(ww * 2 + m) * 16 + row]);
    const int j = j0 + m * 16 + row;
    if (j < cntb) scores[b * Ss + j] = v;
  }
}

// ---------------------------------------------------------------------------
// Stable top-k mask (descending score, ties by index) + scatter to new_mask.
// Sort key s_j = scores_j (real slots have sub_a == 0); padded slots compare
// with s_pad = (bmax + 100*amin) + amin. tokens = max(trunc(f32(valid)*0.8),1)
// where valid counts sub_a == 0 (includes padding iff amin == 0).
// ---------------------------------------------------------------------------
__global__ __launch_bounds__(1024) void topk_mask_kernel(
    const float* __restrict__ scores, const int* __restrict__ comp,
    const int* __restrict__ cnt, const float* __restrict__ ws_f,
    float* __restrict__ out_mask) {
  __shared__ float s_lds[Ss];
  const int b = blockIdx.x;
  const int t = threadIdx.x;
  const int cntb = cnt[b];
  const float amin = ws_f[0];
  const float bmax = ws_f[1];
  for (int i = t; i < Ss; i += 1024) {
    out_mask[b * Ss + i] = 0.f;  // non-compacted positions end up 0
    s_lds[i] = (i < cntb) ? scores[b * Ss + i] : 0.f;
  }
  __syncthreads();
  const float s_pad = (bmax + amin * 100.f) + amin;
  const int validn = cntb + ((amin == 0.f) ? (Ss - cntb) : 0);
  int tokens = (int)((float)validn * KEEP_RATIO);
  if (tokens < 1) tokens = 1;
  for (int j = t; j < cntb; j += 1024) {
    const float sj = s_lds[j];
    int rank = 0;
    for (int i = 0; i < cntb; ++i) {
      const float si = s_lds[i];
      rank += (si > sj) || (si == sj && i < j);
    }
    if (s_pad > sj) rank += Ss - cntb;  // padding slots sort after ties
    out_mask[b * Ss + comp[b * Ss + j]] = (rank < tokens) ? 1.f : 0.f;
  }
}

// ---------------------------------------------------------------------------
extern "C" void kernel_launch(void* const* d_in, const int* in_sizes, int n_in,
                              void* d_out, int out_size, void* d_ws,
                              size_t ws_size, hipStream_t stream) {
  (void)in_sizes; (void)n_in; (void)out_size; (void)ws_size;

  const float* hs    = (const float*)d_in[0];  // [B,S,H]
  const float* amask = (const float*)d_in[1];  // [B,S]
  const float* prev  = (const float*)d_in[2];  // [B,S]
  const float* Wm    = (const float*)d_in[3];  // [H,H]
  const float* bias  = (const float*)d_in[4];  // [H]

  float* out_ret  = (float*)d_out;                       // [B,S,H]
  float* out_mask = out_ret + (size_t)Bb * Ss * Hh;      // [B,S]

  // workspace: [amin, bmax] | cnt[B] | comp_idx[B*S] | scores[B*S]  (~262 KB)
  float* ws_f   = (float*)d_ws;
  int*   cnt    = (int*)((char*)d_ws + 8);
  int*   comp   = (int*)((char*)d_ws + 64);
  float* scores = (float*)((char*)d_ws + 64 + (size_t)Bb * Ss * sizeof(int));

  reduce_minmax_kernel<<<1, 256, 0, stream>>>(amask, bias, ws_f);
  compact_kernel<<<Bb, 1024, 0, stream>>>(prev, comp, cnt);

  const int n4 = (Bb * Ss * Hh) / 4;
  copy_kernel<<<(n4 + 255) / 256, 256, 0, stream>>>((const float4*)hs,
                                                    (float4*)out_ret, n4);

  dim3 grid(Ss / TS_M, Bb);
  const size_t lds_bytes =
      (size_t)(TS_M * A_STRIDE + 8 * 2 * 16) * sizeof(float);
  gemm_scores_kernel<<<grid, 256, lds_bytes, stream>>>(hs, Wm, bias, comp, cnt,
                                                       scores);

  topk_mask_kernel<<<Bb, 1024, 0, stream>>>(scores, comp, cnt, ws_f, out_mask);
}